// AttentionHead_64587718197641
// MI455X (gfx1250) — compile-verified
//
#include <hip/hip_runtime.h>
#include <hip/hip_bf16.h>

// CDNA5 (gfx1250) fused single-head attention, full f32 via V_WMMA_F32_16X16X4_F32.
// B=8, S=2048, WIDTH=1024, HEAD=64.

typedef __attribute__((ext_vector_type(2))) float v2f;
typedef __attribute__((ext_vector_type(4))) float v4f;
typedef __attribute__((ext_vector_type(8))) float v8f;

#define WIDTH 1024
#define HEAD  64
#define NB    8
#define SEQ   2048
#define MROWS (NB * SEQ)   // 16384

__device__ __forceinline__ v8f wmma_f32(v2f a, v2f b, v8f c) {
  // (neg_a, A, neg_b, B, c_mod, C, reuse_a, reuse_b)
  return __builtin_amdgcn_wmma_f32_16x16x4_f32(
      false, a, false, b, (short)0, c, false, false);
}

// ---------------------------------------------------------------------------
// Kernel 1: q/k/v = x * W + b.  One wave computes a 32-row x 64-col tile of
// q, k and v simultaneously: 2 A fragments (row groups) reuse each W B
// fragment twice -> 24 WMMA per ~26 loads per K-step.
// A layout (16x4 f32): lane L holds row M=L&15, K = (L>>4)*2 + j.
// B layout (4x16 f32): lane L holds col N=L&15, K = (L>>4)*2 + j.
// ---------------------------------------------------------------------------
__global__ __launch_bounds__(256) void qkv_proj_kernel(
    const float* __restrict__ x,
    const float* __restrict__ Wq, const float* __restrict__ bq,
    const float* __restrict__ Wk, const float* __restrict__ bk,
    const float* __restrict__ Wv, const float* __restrict__ bv,
    float* __restrict__ qout, float* __restrict__ kout, float* __restrict__ vout)
{
  const int lane = threadIdx.x & 31;
  const int wave = threadIdx.x >> 5;
  const int tile = blockIdx.x * 8 + wave;      // 0..511 (32-row tiles)
  const int row0 = tile * 32;
  const int l15  = lane & 15;
  const int hh   = lane >> 4;                  // half-wave id (0/1)

  v8f accq[2][4], acck[2][4], accv[2][4];
  #pragma unroll
  for (int g = 0; g < 2; ++g)
    #pragma unroll
    for (int t = 0; t < 4; ++t) { accq[g][t] = {}; acck[g][t] = {}; accv[g][t] = {}; }

  const float* xrow0 = x + (size_t)(row0 + l15) * WIDTH + hh * 2;
  const float* xrow1 = xrow0 + 16 * WIDTH;

  for (int k0 = 0; k0 < WIDTH; k0 += 4) {
    v2f a0 = *(const v2f*)(xrow0 + k0);        // x[row     , k0+hh*2+{0,1}]
    v2f a1 = *(const v2f*)(xrow1 + k0);        // x[row + 16, ...]
    __builtin_prefetch(xrow0 + k0 + 128, 0, 3);
    __builtin_prefetch(xrow1 + k0 + 128, 0, 3);
    const int kr = k0 + hh * 2;
    #pragma unroll
    for (int t = 0; t < 4; ++t) {
      const int col = t * 16 + l15;
      v2f bw;
      bw[0] = Wq[(size_t)kr * HEAD + col];
      bw[1] = Wq[(size_t)(kr + 1) * HEAD + col];
      accq[0][t] = wmma_f32(a0, bw, accq[0][t]);
      accq[1][t] = wmma_f32(a1, bw, accq[1][t]);
      bw[0] = Wk[(size_t)kr * HEAD + col];
      bw[1] = Wk[(size_t)(kr + 1) * HEAD + col];
      acck[0][t] = wmma_f32(a0, bw, acck[0][t]);
      acck[1][t] = wmma_f32(a1, bw, acck[1][t]);
      bw[0] = Wv[(size_t)kr * HEAD + col];
      bw[1] = Wv[(size_t)(kr + 1) * HEAD + col];
      accv[0][t] = wmma_f32(a0, bw, accv[0][t]);
      accv[1][t] = wmma_f32(a1, bw, accv[1][t]);
    }
  }

  // C/D layout: VGPR r -> row (r + hh*8), lane -> col (l15).
  #pragma unroll
  for (int t = 0; t < 4; ++t) {
    const int col = t * 16 + l15;
    const float biq = bq[col], bik = bk[col], biv = bv[col];
    #pragma unroll
    for (int g = 0; g < 2; ++g) {
      #pragma unroll
      for (int r = 0; r < 8; ++r) {
        const size_t idx = (size_t)(row0 + g * 16 + r + hh * 8) * HEAD + col;
        qout[idx] = accq[g][t][r] + biq;
        kout[idx] = acck[g][t][r] + bik;
        vout[idx] = accv[g][t][r] + biv;
      }
    }
  }
}

// ---------------------------------------------------------------------------
// Kernel 2: flash attention. One wave owns 32 queries (two 16-query groups),
// loops over 16-key blocks. Computes S' = K.Q^T (keys on M axis) so softmax
// reductions are within-lane over VGPRs + one shfl_xor(16), and P^T (C/D
// layout) converts to the B operand of O^T = V^T . P^T with 8 shfl_xor(16)
// per group per block (no LDS). K/V fragments are shared by both query
// groups -> 64 WMMA per ~50 loads per key block.
// ---------------------------------------------------------------------------
__global__ __launch_bounds__(256) void flash_attn_kernel(
    const float* __restrict__ q, const float* __restrict__ k,
    const float* __restrict__ v, const int* __restrict__ mask,
    float* __restrict__ out)
{
  const int lane  = threadIdx.x & 31;
  const int wave  = threadIdx.x >> 5;
  const int qtile = blockIdx.x * 8 + wave;     // 0..511 (32-query tiles)
  const int batch = qtile >> 6;                // 64 tiles per batch
  const int qBase = qtile * 32;                // global row (includes batch)
  const int l15   = lane & 15;
  const int hh    = lane >> 4;
  const bool lo   = (hh == 0);

  // Preload Q as B fragments: 2 groups x 16 h-steps (h = 4*s + hh*2 + j).
  v2f qf[2][16];
  #pragma unroll
  for (int g = 0; g < 2; ++g) {
    const float* qrow = q + (size_t)(qBase + g * 16 + l15) * HEAD + hh * 2;
    #pragma unroll
    for (int s = 0; s < 16; ++s) qf[g][s] = *(const v2f*)(qrow + s * 4);
  }

  v8f o[2][4];
  #pragma unroll
  for (int g = 0; g < 2; ++g)
    #pragma unroll
    for (int t = 0; t < 4; ++t) o[g][t] = {};
  float l_run[2] = {0.0f, 0.0f};
  float m_run[2] = {-INFINITY, -INFINITY};

  const int keyStart = batch * SEQ;
  const float* kbp = k + (size_t)keyStart * HEAD;
  const float* vbp = v + (size_t)keyStart * HEAD;
  const int*   mbp = mask + keyStart;

  for (int kb0 = 0; kb0 < SEQ; kb0 += 16) {
    // ---- scores: S'(key, q), 16 keys x (2x16) queries, K-dim = HEAD
    v8f s[2]; s[0] = {}; s[1] = {};
    const float* krow = kbp + (size_t)(kb0 + l15) * HEAD + hh * 2;
    #pragma unroll
    for (int hs = 0; hs < 16; ++hs) {
      v2f af = *(const v2f*)(krow + hs * 4);   // K rows as A fragment (shared)
      s[0] = wmma_f32(af, qf[0][hs], s[0]);
      s[1] = wmma_f32(af, qf[1][hs], s[1]);
    }

    const int maskv = mbp[kb0 + l15];          // lane i holds mask[key i&15]
    v2f pb[2][4];

    #pragma unroll
    for (int g = 0; g < 2; ++g) {
      // ---- scale + key-padding mask. key = r + hh*8 for VGPR r.
      float p[8];
      float tmax = -INFINITY;
      #pragma unroll
      for (int r = 0; r < 8; ++r) {
        float sv = s[g][r] * 0.125f;           // 1/sqrt(64)
        const int mr = __shfl(maskv, r + hh * 8, 32);
        sv = (mr == 0) ? -INFINITY : sv;
        p[r] = sv;
        tmax = fmaxf(tmax, sv);
      }
      tmax = fmaxf(tmax, __shfl_xor(tmax, 16, 32));  // combine key halves

      const float m_new = fmaxf(m_run[g], tmax);
      const float alpha = (m_run[g] == -INFINITY) ? 0.0f : __expf(m_run[g] - m_new);

      float psum = 0.0f;
      #pragma unroll
      for (int r = 0; r < 8; ++r) {
        const float e = (p[r] == -INFINITY) ? 0.0f : __expf(p[r] - m_new);
        p[r] = e;
        psum += e;
      }
      psum += __shfl_xor(psum, 16, 32);
      l_run[g] = l_run[g] * alpha + psum;
      m_run[g] = m_new;
      #pragma unroll
      for (int t = 0; t < 4; ++t)
        #pragma unroll
        for (int r = 0; r < 8; ++r) o[g][t][r] *= alpha;

      // ---- P^T (C/D layout) -> B fragments (K=key, N=query) per kk step.
      float ps[8];
      #pragma unroll
      for (int r = 0; r < 8; ++r) ps[r] = __shfl_xor(p[r], 16, 32);
      pb[g][0][0] = lo ? p[0]  : ps[2];  pb[g][0][1] = lo ? p[1]  : ps[3];
      pb[g][1][0] = lo ? p[4]  : ps[6];  pb[g][1][1] = lo ? p[5]  : ps[7];
      pb[g][2][0] = lo ? ps[0] : p[2];   pb[g][2][1] = lo ? ps[1] : p[3];
      pb[g][3][0] = lo ? ps[4] : p[6];   pb[g][3][1] = lo ? ps[5] : p[7];
    }

    // ---- O^T(h, q) += V^T(h, key) . P^T(key, q); V fragments shared.
    #pragma unroll
    for (int kk = 0; kk < 4; ++kk) {
      const float* vrow = vbp + (size_t)(kb0 + kk * 4 + hh * 2) * HEAD + l15;
      #pragma unroll
      for (int t = 0; t < 4; ++t) {
        v2f av;
        av[0] = vrow[t * 16];                 // key = kk*4 + hh*2
        av[1] = vrow[HEAD + t * 16];          // key = kk*4 + hh*2 + 1
        o[0][t] = wmma_f32(av, pb[0][kk], o[0][t]);
        o[1][t] = wmma_f32(av, pb[1][kk], o[1][t]);
      }
    }
  }

  // ---- normalize and store. O^T tile t: VGPR r -> h = t*16 + r + hh*8,
  // lane -> q = l15. 8 consecutive h per lane -> two b128 stores per tile.
  #pragma unroll
  for (int g = 0; g < 2; ++g) {
    const float inv_l = 1.0f / l_run[g];
    #pragma unroll
    for (int t = 0; t < 4; ++t) {
      v4f loq, hiq;
      #pragma unroll
      for (int r = 0; r < 4; ++r) {
        loq[r] = o[g][t][r] * inv_l;
        hiq[r] = o[g][t][r + 4] * inv_l;
      }
      float* op = out + (size_t)(qBase + g * 16 + l15) * HEAD + t * 16 + hh * 8;
      *(v4f*)op       = loq;
      *((v4f*)op + 1) = hiq;
    }
  }
}

// ---------------------------------------------------------------------------
extern "C" void kernel_launch(void* const* d_in, const int* in_sizes, int n_in,
                              void* d_out, int out_size, void* d_ws, size_t ws_size,
                              hipStream_t stream) {
  const float* x    = (const float*)d_in[0];
  const int*   mask = (const int*)d_in[1];
  const float* Wq   = (const float*)d_in[2];
  const float* bq   = (const float*)d_in[3];
  const float* Wk   = (const float*)d_in[4];
  const float* bk   = (const float*)d_in[5];
  const float* Wv   = (const float*)d_in[6];
  const float* bv   = (const float*)d_in[7];
  float* out = (float*)d_out;

  // Workspace: q,k,v staging, 3 * 16384*64 f32 = 12.6 MB (fully overwritten
  // every call -> deterministic).
  float* qbuf = (float*)d_ws;
  float* kbuf = qbuf + (size_t)MROWS * HEAD;
  float* vbuf = kbuf + (size_t)MROWS * HEAD;

  // 512 row-tiles of 32, 8 waves (256 thr) per block -> 64 blocks each.
  qkv_proj_kernel<<<64, 256, 0, stream>>>(x, Wq, bq, Wk, bk, Wv, bv,
                                          qbuf, kbuf, vbuf);
  flash_attn_kernel<<<64, 256, 0, stream>>>(qbuf, kbuf, vbuf, mask, out);
}